// DetectionLoss_37014028157028
// MI455X (gfx1250) — compile-verified
//
#include <hip/hip_runtime.h>

#define B_     32
#define M_     64
#define GRID_  92
#define A_     (GRID_ * GRID_)      // 8464
#define NC_    80
#define RM_    16
#define TOPK_  10
#define EPS_   1e-7f
#define NPART_ ((B_ * A_ + 255) / 256)   // 1058, exact: 256*1058 == B_*A_

static_assert(256 * NPART_ == B_ * A_, "grid must tile exactly");

typedef float v2f __attribute__((ext_vector_type(2)));
typedef float v8f __attribute__((ext_vector_type(8)));

// ---------------------------------------------------------------------------
// Kernel 1: TaskAlignedAssigner. One block per (b, m) GT slot.
// Computes align_metric over all anchors into LDS, extracts top-10
// (ties -> lower index, matching jax.lax.top_k), applies mask_gt & in_gt,
// scatters candidates into a per-(b,a) argmax table via 64-bit atomicMax
// (packed: [value bits | M-m] so larger value wins, ties -> smaller m,
//  matching jnp.argmax first-max semantics), and records max_per_gt.
// ---------------------------------------------------------------------------
__global__ __launch_bounds__(256) void assign_kernel(
    const float* __restrict__ pred_scores,
    const float* __restrict__ pred_bboxes,
    const float* __restrict__ anchors,
    const int*   __restrict__ gt_labels,
    const float* __restrict__ gt_bboxes,
    const float* __restrict__ mask_gt,
    unsigned long long* __restrict__ argmax_tab,
    float* __restrict__ maxpergt)
{
    __shared__ float smetric[A_];       // 33856 B (WGP has 320 KB LDS)
    __shared__ float rval[256];
    __shared__ int   ridx[256];
    __shared__ int   selidx[TOPK_];
    __shared__ float selval[TOPK_];
    __shared__ float cand[TOPK_];

    const int bm  = blockIdx.x;
    const int b   = bm / M_;
    const int m   = bm % M_;
    const int tid = threadIdx.x;

    const float4 g  = ((const float4*)gt_bboxes)[bm];
    const float gx1 = g.x, gy1 = g.y, gx2 = g.z, gy2 = g.w;
    int cls = gt_labels[bm];
    cls = cls < 0 ? 0 : (cls > NC_ - 1 ? NC_ - 1 : cls);
    const float mgt  = mask_gt[bm];
    const float a_gt = (gx2 - gx1) * (gy2 - gy1);

    const float4* __restrict__ pb = (const float4*)pred_bboxes + (size_t)b * A_;
    const float*  __restrict__ sc = pred_scores + (size_t)b * A_ * NC_;

    for (int a = tid; a < A_; a += 256) {
        float4 p  = pb[a];
        float iw  = fmaxf(fminf(gx2, p.z) - fmaxf(gx1, p.x), 0.f);
        float ih  = fmaxf(fminf(gy2, p.w) - fmaxf(gy1, p.y), 0.f);
        float inter = iw * ih;
        float a_pd  = (p.z - p.x) * (p.w - p.y);
        float ov    = inter / (a_gt + a_pd - inter + EPS_);
        float s     = sc[(size_t)a * NC_ + cls];
        float sig   = 1.f / (1.f + expf(-s));
        float p2 = ov * ov;
        smetric[a] = sqrtf(sig) * (p2 * p2 * p2) * mgt;   // sig^0.5 * ov^6 * mask
    }
    __syncthreads();

    // top-10 by repeated block argmax (ties -> lower index)
    for (int k = 0; k < TOPK_; ++k) {
        float bv = -1.f; int bi = 0x7fffffff;
        for (int a = tid; a < A_; a += 256) {
            float v = smetric[a];
            if (v > bv || (v == bv && a < bi)) { bv = v; bi = a; }
        }
        rval[tid] = bv; ridx[tid] = bi;
        __syncthreads();
        for (int s = 128; s > 0; s >>= 1) {
            if (tid < s) {
                float ov_ = rval[tid + s]; int oi = ridx[tid + s];
                if (ov_ > rval[tid] || (ov_ == rval[tid] && oi < ridx[tid])) {
                    rval[tid] = ov_; ridx[tid] = oi;
                }
            }
            __syncthreads();
        }
        if (tid == 0) {
            selidx[k] = ridx[0];
            selval[k] = rval[0];
            smetric[ridx[0]] = -2.f;   // remove from further rounds
        }
        __syncthreads();
    }

    if (tid < TOPK_) {
        int a   = selidx[tid];
        float v = selval[tid];
        float2 an = ((const float2*)anchors)[a];
        float dmin = fminf(fminf(an.x - gx1, an.y - gy1),
                           fminf(gx2 - an.x, gy2 - an.y));
        bool ok = (mgt > 0.f) && (dmin > EPS_);
        cand[tid] = ok ? v : 0.f;
        if (ok) {
            unsigned long long pk =
                ((unsigned long long)__float_as_uint(v) << 32) |
                (unsigned long long)(unsigned)(M_ - m);
            atomicMax(&argmax_tab[(size_t)b * A_ + a], pk);
        }
    }
    __syncthreads();
    if (tid == 0) {
        float mx = 0.f;
        for (int k = 0; k < TOPK_; ++k) mx = fmaxf(mx, cand[k]);
        maxpergt[bm] = mx;
    }
}

// ---------------------------------------------------------------------------
// Kernel 2: per-anchor loss. One thread per (b,a). BCE over all 80 classes;
// CIoU and DFL only for foreground anchors (skips ~99% of pred_dist reads).
// Deterministic block tree-reduction of 5 partial sums.
// ---------------------------------------------------------------------------
__global__ __launch_bounds__(256) void loss_kernel(
    const float* __restrict__ pred_scores,
    const float* __restrict__ pred_dist,
    const float* __restrict__ pred_bboxes,
    const float* __restrict__ anchors,
    const int*   __restrict__ gt_labels,
    const float* __restrict__ gt_bboxes,
    const unsigned long long* __restrict__ argmax_tab,
    const float* __restrict__ maxpergt,
    float* __restrict__ part)
{
    const int tid = threadIdx.x;
    const int i   = blockIdx.x * 256 + tid;   // exact tiling, no guard needed
    const int b   = i / A_;
    const int a   = i % A_;

    float bce = 0.f, tsc = 0.f, iouL = 0.f, dflL = 0.f, nfg = 0.f;

    unsigned long long pk = argmax_tab[i];
    const bool fg = (pk != 0ull);
    float ns = 0.f; int tlab = 0;
    float tb1 = 0.f, tb2 = 0.f, tb3 = 0.f, tb4 = 0.f;
    if (fg) {
        int m    = M_ - (int)(unsigned)(pk & 0xffffffffull);
        float am = __uint_as_float((unsigned)(pk >> 32));
        ns       = am / (maxpergt[b * M_ + m] + EPS_);
        int cls  = gt_labels[b * M_ + m];
        tlab     = cls < 0 ? 0 : (cls > NC_ - 1 ? NC_ - 1 : cls);
        float4 g = ((const float4*)gt_bboxes)[b * M_ + m];
        tb1 = g.x; tb2 = g.y; tb3 = g.z; tb4 = g.w;
        nfg = 1.f; tsc = ns;
    }

    // BCE: sum_c [max(x,0) + log1p(exp(-|x|))] - x[tlab]*ns   (one-hot target)
    const float4* __restrict__ xs = (const float4*)(pred_scores + (size_t)i * NC_);
    float s0 = 0.f;
    #pragma unroll 4
    for (int c4 = 0; c4 < NC_ / 4; ++c4) {
        float4 x = xs[c4];
        s0 += fmaxf(x.x, 0.f) + log1pf(expf(-fabsf(x.x)));
        s0 += fmaxf(x.y, 0.f) + log1pf(expf(-fabsf(x.y)));
        s0 += fmaxf(x.z, 0.f) + log1pf(expf(-fabsf(x.z)));
        s0 += fmaxf(x.w, 0.f) + log1pf(expf(-fabsf(x.w)));
    }
    bce = s0;
    if (fg) bce -= pred_scores[(size_t)i * NC_ + tlab] * ns;

    if (fg) {
        // ------ CIoU ------
        float4 p = ((const float4*)pred_bboxes)[i];
        float x11 = p.x, y11 = p.y, x21 = p.z, y21 = p.w;
        float x12 = tb1, y12 = tb2, x22 = tb3, y22 = tb4;
        float iw = fmaxf(fminf(x21, x22) - fmaxf(x11, x12), 0.f);
        float ih = fmaxf(fminf(y21, y22) - fmaxf(y11, y12), 0.f);
        float inter = iw * ih;
        float a1 = (x21 - x11) * (y21 - y11);
        float a2 = (x22 - x12) * (y22 - y12);
        float iou = inter / (a1 + a2 - inter + EPS_);
        float cw = fmaxf(x21, x22) - fminf(x11, x12);
        float ch = fmaxf(y21, y22) - fminf(y11, y12);
        float c2 = cw * cw + ch * ch + EPS_;
        float dx = x11 + x21 - x12 - x22;
        float dy = y11 + y21 - y12 - y22;
        float rho2 = (dx * dx + dy * dy) * 0.25f;
        float w1 = x21 - x11, h1 = y21 - y11, w2 = x22 - x12, h2 = y22 - y12;
        const float pif = 3.14159265358979323846f;
        float da = atanf(w1 / (h1 + EPS_)) - atanf(w2 / (h2 + EPS_));
        float v  = (4.0f / (pif * pif)) * da * da;
        float alpha = v / (1.f - iou + v + EPS_);
        float ciou  = iou - (rho2 / c2 + v * alpha);
        iouL = (1.f - ciou) * ns;

        // ------ DFL ------
        float2 an = ((const float2*)anchors)[a];
        float tt4[4] = { an.x - tb1, an.y - tb2, tb3 - an.x, tb4 - an.y };
        const float4* __restrict__ pd = (const float4*)(pred_dist + (size_t)i * 4 * RM_);
        for (int sd = 0; sd < 4; ++sd) {
            float lg[RM_];
            float4 q;
            q = pd[sd * 4 + 0]; lg[0]  = q.x; lg[1]  = q.y; lg[2]  = q.z; lg[3]  = q.w;
            q = pd[sd * 4 + 1]; lg[4]  = q.x; lg[5]  = q.y; lg[6]  = q.z; lg[7]  = q.w;
            q = pd[sd * 4 + 2]; lg[8]  = q.x; lg[9]  = q.y; lg[10] = q.z; lg[11] = q.w;
            q = pd[sd * 4 + 3]; lg[12] = q.x; lg[13] = q.y; lg[14] = q.z; lg[15] = q.w;
            float mx = lg[0];
            #pragma unroll
            for (int j = 1; j < RM_; ++j) mx = fmaxf(mx, lg[j]);
            float se = 0.f;
            #pragma unroll
            for (int j = 0; j < RM_; ++j) se += expf(lg[j] - mx);
            float logZ = mx + logf(se);
            float t  = fminf(fmaxf(tt4[sd], 0.f), (float)RM_ - 1.f - 0.01f);
            int   tl = (int)t;
            int   tr = (tl + 1 > RM_ - 1) ? RM_ - 1 : tl + 1;
            float wl = (float)(tl + 1) - t;
            float wr = 1.f - wl;
            dflL += (logZ - lg[tl]) * wl + (logZ - lg[tr]) * wr;
        }
    }

    // deterministic block reduction of the five sums
    __shared__ float sred[256];
    float vals[5] = { bce, tsc, iouL, dflL, nfg };
    for (int q = 0; q < 5; ++q) {
        sred[tid] = vals[q];
        __syncthreads();
        for (int s = 128; s > 0; s >>= 1) {
            if (tid < s) sred[tid] += sred[tid + s];
            __syncthreads();
        }
        if (tid == 0) part[q * NPART_ + blockIdx.x] = sred[0];
        __syncthreads();
    }
}

// ---------------------------------------------------------------------------
// Kernel 3: final reduction + loss combine. One wave (EXEC all ones, as WMMA
// requires). Uses V_WMMA_F32_16X16X4_F32 with B = ones as an fp32 reduction
// engine: each issue folds 64 partials into the accumulator (row sums,
// replicated over 16 columns -> divide by 16, exact in fp32).
// ---------------------------------------------------------------------------
__global__ __launch_bounds__(32) void finalize_kernel(
    const float* __restrict__ part, float* __restrict__ out)
{
    const int lane = threadIdx.x;
    float tot[5];
    v2f ones; ones.x = 1.f; ones.y = 1.f;

    for (int q = 0; q < 5; ++q) {
        v8f c = { 0.f, 0.f, 0.f, 0.f, 0.f, 0.f, 0.f, 0.f };
        for (int base = 0; base < NPART_; base += 64) {
            int i0 = base + lane * 2;
            int i1 = i0 + 1;
            v2f aa;
            aa.x = (i0 < NPART_) ? part[q * NPART_ + i0] : 0.f;
            aa.y = (i1 < NPART_) ? part[q * NPART_ + i1] : 0.f;
            // D = A x ones + C : accumulates 64-element row sums in fp32
            c = __builtin_amdgcn_wmma_f32_16x16x4_f32(
                    false, aa, false, ones, (short)0, c, false, false);
        }
        float s = c[0] + c[1] + c[2] + c[3] + c[4] + c[5] + c[6] + c[7];
        for (int off = 16; off > 0; off >>= 1)
            s += __shfl_xor(s, off, 32);
        tot[q] = s * (1.0f / 16.0f);   // column replication factor (exact)
    }

    if (lane == 0) {
        float tss      = fmaxf(tot[1], 1.0f);
        float loss_cls = tot[0] / tss;
        float loss_iou = tot[2] / tss;
        float nfg      = tot[4];
        float loss_dfl = (nfg > 0.f) ? tot[3] / fmaxf(4.f * nfg, 1.f) : 0.f;
        out[0] = 7.5f * loss_iou + 0.5f * loss_cls + 1.5f * loss_dfl;
    }
}

// ---------------------------------------------------------------------------
extern "C" void kernel_launch(void* const* d_in, const int* in_sizes, int n_in,
                              void* d_out, int out_size, void* d_ws, size_t ws_size,
                              hipStream_t stream)
{
    (void)in_sizes; (void)n_in; (void)out_size; (void)ws_size;

    const float* pred_scores = (const float*)d_in[0];
    const float* pred_dist   = (const float*)d_in[1];
    const float* pred_bboxes = (const float*)d_in[2];
    const float* anchors     = (const float*)d_in[3];
    const int*   gt_labels   = (const int*)d_in[4];
    const float* gt_bboxes   = (const float*)d_in[5];
    const float* mask_gt     = (const float*)d_in[6];
    float*       out         = (float*)d_out;

    char* ws = (char*)d_ws;
    unsigned long long* argmax_tab = (unsigned long long*)ws;        // B*A u64
    size_t off0 = (size_t)B_ * A_ * sizeof(unsigned long long);
    float* maxpergt = (float*)(ws + off0);                           // B*M f32
    size_t off1 = off0 + (size_t)B_ * M_ * sizeof(float);
    float* part = (float*)(ws + off1);                               // 5*NPART f32

    hipMemsetAsync(argmax_tab, 0, (size_t)B_ * A_ * sizeof(unsigned long long), stream);

    assign_kernel<<<B_ * M_, 256, 0, stream>>>(
        pred_scores, pred_bboxes, anchors, gt_labels, gt_bboxes, mask_gt,
        argmax_tab, maxpergt);

    loss_kernel<<<NPART_, 256, 0, stream>>>(
        pred_scores, pred_dist, pred_bboxes, anchors, gt_labels, gt_bboxes,
        argmax_tab, maxpergt, part);

    finalize_kernel<<<1, 32, 0, stream>>>(part, out);
}